// HistogramPooling_89498528514147
// MI455X (gfx1250) — compile-verified
//
#include <hip/hip_runtime.h>

#define BINS 64
#define HW_ELEMS 4096      // 64*64 elements per (B,C) slice
#define TPB 256            // 8 wave32
#define WAVES 8
// each thread owns 16 elements = 4 x float4

__global__ __launch_bounds__(TPB) void histc_slice_kernel(const float* __restrict__ x,
                                                          float* __restrict__ out) {
    __shared__ float    sdata[HW_ELEMS];        // 16 KB slice staging
    __shared__ unsigned whist[WAVES][BINS];     // 2 KB per-wave privatized histograms
    __shared__ float    wmin[WAVES];
    __shared__ float    wmax[WAVES];

    const int tid   = threadIdx.x;
    const int wave  = tid >> 5;
    const int slice = blockIdx.x;
    const float* gbase = x + (size_t)slice * HW_ELEMS;

    // ---- Stage 1: async copy global -> LDS (CDNA5 async path, ASYNCcnt) ----
    // thread t covers float4 slots { t, t+256, t+512, t+768 }.
    // INST_OFFSET is added to BOTH the LDS address and the global address,
    // so one base pair + 4 immediate offsets covers the whole slice.
    {
        unsigned ldsAddr   = (unsigned)(uintptr_t)(&sdata[tid * 4]);
        const float* gaddr = gbase + tid * 4;
        asm volatile("global_load_async_to_lds_b128 %0, %1, off"
                     :: "v"(ldsAddr), "v"(gaddr) : "memory");
        asm volatile("global_load_async_to_lds_b128 %0, %1, off offset:4096"
                     :: "v"(ldsAddr), "v"(gaddr) : "memory");
        asm volatile("global_load_async_to_lds_b128 %0, %1, off offset:8192"
                     :: "v"(ldsAddr), "v"(gaddr) : "memory");
        asm volatile("global_load_async_to_lds_b128 %0, %1, off offset:12288"
                     :: "v"(ldsAddr), "v"(gaddr) : "memory");
    }

    // Zero per-wave histograms while the async copy is in flight (512 uints).
    {
        unsigned* wh = &whist[0][0];
        wh[tid]       = 0u;
        wh[tid + 256] = 0u;
    }

    asm volatile("s_wait_asynccnt 0" ::: "memory");
    __syncthreads();

    // ---- Stage 2: per-thread min/max over its 16 staged elements ----
    const float4* sv = reinterpret_cast<const float4*>(sdata);
    float4 a0 = sv[tid];
    float4 a1 = sv[tid + 256];
    float4 a2 = sv[tid + 512];
    float4 a3 = sv[tid + 768];

    float vals[16] = {a0.x, a0.y, a0.z, a0.w,
                      a1.x, a1.y, a1.z, a1.w,
                      a2.x, a2.y, a2.z, a2.w,
                      a3.x, a3.y, a3.z, a3.w};

    float mn = vals[0], mx = vals[0];
#pragma unroll
    for (int i = 1; i < 16; ++i) {
        mn = fminf(mn, vals[i]);
        mx = fmaxf(mx, vals[i]);
    }

    // wave32 butterfly reduction
#pragma unroll
    for (int off = 16; off >= 1; off >>= 1) {
        mn = fminf(mn, __shfl_xor(mn, off, 32));
        mx = fmaxf(mx, __shfl_xor(mx, off, 32));
    }
    if ((tid & 31) == 0) { wmin[wave] = mn; wmax[wave] = mx; }
    __syncthreads();

    float gmn = wmin[0], gmx = wmax[0];
#pragma unroll
    for (int w = 1; w < WAVES; ++w) {
        gmn = fminf(gmn, wmin[w]);
        gmx = fmaxf(gmx, wmax[w]);
    }
    // torch.histc semantics: degenerate range widened by +/-1
    if (gmn == gmx) { gmn -= 1.0f; gmx += 1.0f; }
    const float range = gmx - gmn;

    // ---- Stage 3: bin from registers into per-wave LDS histograms ----
    unsigned* myhist = whist[wave];
#pragma unroll
    for (int i = 0; i < 16; ++i) {
        // match reference arithmetic order: (v - mn) / (mx - mn) * bins
        int idx = (int)floorf((vals[i] - gmn) / range * 64.0f);
        idx = idx < 0 ? 0 : (idx > BINS - 1 ? BINS - 1 : idx);
        atomicAdd(&myhist[idx], 1u);   // ds_add_u32
    }
    __syncthreads();

    // ---- Stage 4: fold 8 partials, coalesced f32 store (256 B per slice) ----
    if (tid < BINS) {
        unsigned s = 0;
#pragma unroll
        for (int w = 0; w < WAVES; ++w) s += whist[w][tid];
        out[(size_t)slice * BINS + tid] = (float)s;
    }
}

extern "C" void kernel_launch(void* const* d_in, const int* in_sizes, int n_in,
                              void* d_out, int out_size, void* d_ws, size_t ws_size,
                              hipStream_t stream) {
    const float* x = (const float*)d_in[0];
    float* out     = (float*)d_out;
    // num_of_bins (d_in[1]) is the compile-time constant 64 in the reference.
    const int n_slices = in_sizes[0] / HW_ELEMS;   // 64*256 = 16384
    histc_slice_kernel<<<n_slices, TPB, 0, stream>>>(x, out);
}